// WaveletLayer02_68367289418120
// MI455X (gfx1250) — compile-verified
//
#include <hip/hip_runtime.h>
#include <hip/hip_bf16.h>

typedef __attribute__((ext_vector_type(2))) float v2f;
typedef __attribute__((ext_vector_type(8))) float v8f;

#define Bsz 2048
#define Nsz 512
#define Dsz 128
#define BT 32          // output tile rows  (b)
#define NT 32          // output tile cols  (n)
#define LDP (BT + 4)   // padded row stride in LDS (36 floats = 144B, 16B aligned)

// h[b,n] = prod_d (1 - z^2) * exp(-0.5 * sum_d z^2),  z = x*r - br, r = 1/scale, br = bias/scale
// sum_d z^2 = (X^2 @ (R^2)^T) - 2 (X @ (R*BR)^T) + c_n   -> fp32 WMMA GEMM (K=256)
// prod term -> register-blocked VALU loop (3 ops/element: fma, mul, fma), fused exp at store.
__global__ __launch_bounds__(128, 1)
void WaveletLayer02_68367289418120_kernel(const float* __restrict__ x,
                                          const float* __restrict__ bias,
                                          const float* __restrict__ scale,
                                          float* __restrict__ out)
{
    __shared__ float xsT[Dsz][LDP];   // x tile, transposed: [d][row]
    __shared__ float rsT[Dsz][LDP];   // r  = 1/scale, transposed: [d][col]
    __shared__ float brT[Dsz][LDP];   // br = bias/scale, transposed: [d][col]
    __shared__ float cs4[4][NT];      // partial c_n sums
    __shared__ float cs[NT];          // c_n = sum_d br^2
    __shared__ float sS[BT][NT];      // S = sum_d z^2 per tile element

    const int tid  = threadIdx.x;     // 0..127
    const int wave = tid >> 5;        // 0..3
    const int lane = tid & 31;
    const int b0   = blockIdx.x * BT;
    const int n0   = blockIdx.y * NT;

    // ---------------- stage tiles into LDS (transposed) ----------------
    {
        const int r  = tid >> 2;          // 0..31 : row of x tile / col of bias tile
        const int c0 = (tid & 3) * 32;    // 0,32,64,96 : d-quarter
        const float* xrow = x     + (size_t)(b0 + r) * Dsz + c0;
        const float* brow = bias  + (size_t)(n0 + r) * Dsz + c0;
        const float* srow = scale + (size_t)(n0 + r) * Dsz + c0;
        #pragma unroll
        for (int i = 0; i < 32; i += 4) {
            const float4 xv = *(const float4*)(xrow + i);
            const float4 bv = *(const float4*)(brow + i);
            const float4 sv = *(const float4*)(srow + i);
            const int d = c0 + i;
            const float r0 = 1.0f / sv.x, r1 = 1.0f / sv.y;
            const float r2 = 1.0f / sv.z, r3 = 1.0f / sv.w;
            xsT[d+0][r] = xv.x;      xsT[d+1][r] = xv.y;
            xsT[d+2][r] = xv.z;      xsT[d+3][r] = xv.w;
            rsT[d+0][r] = r0;        rsT[d+1][r] = r1;
            rsT[d+2][r] = r2;        rsT[d+3][r] = r3;
            brT[d+0][r] = bv.x * r0; brT[d+1][r] = bv.y * r1;
            brT[d+2][r] = bv.z * r2; brT[d+3][r] = bv.w * r3;
        }
    }
    __syncthreads();

    // ---------------- c_n = sum_d br^2 (all 128 threads, 4-way split over d) ----------------
    {
        const int n  = tid & 31;
        const int dq = (tid >> 5) * 32;
        float c = 0.0f;
        #pragma unroll 8
        for (int d = dq; d < dq + 32; ++d) {
            const float bv = brT[d][n];
            c = fmaf(bv, bv, c);
        }
        cs4[tid >> 5][n] = c;
    }
    __syncthreads();
    if (tid < NT) cs[tid] = cs4[0][tid] + cs4[1][tid] + cs4[2][tid] + cs4[3][tid];
    __syncthreads();

    // ---------------- WMMA phase: S = X^2 @ (R^2)^T - 2 X @ (R*BR)^T ----------------
    // Each wave owns one 16x16 subtile of the 32x32 S tile.
    {
        const int wr = (wave >> 1) * 16;       // subtile row base
        const int wc = (wave & 1) * 16;        // subtile col base
        const int ml = wr + (lane & 15);       // A-fragment M per lane
        const int nl = wc + (lane & 15);       // B-fragment N per lane
        const int kh = (lane >> 4) * 2;        // K sub-offset: lanes 16-31 -> +2

        v8f acc = {};
        #pragma unroll 4
        for (int k0 = 0; k0 < Dsz; k0 += 4) {
            const int d0 = k0 + kh, d1 = d0 + 1;
            const float xa0 = xsT[d0][ml], xa1 = xsT[d1][ml];
            const float r0  = rsT[d0][nl], r1  = rsT[d1][nl];
            const float br0 = brT[d0][nl], br1 = brT[d1][nl];
            v2f a, bm;
            // pass 1: A = x^2, B = r^2
            a.x = xa0 * xa0;  a.y = xa1 * xa1;
            bm.x = r0 * r0;   bm.y = r1 * r1;
            acc = __builtin_amdgcn_wmma_f32_16x16x4_f32(false, a, false, bm,
                                                        (short)0, acc, false, false);
            // pass 2: A = x, B = -2*r*br
            a.x = xa0;                a.y = xa1;
            bm.x = -2.0f * r0 * br0;  bm.y = -2.0f * r1 * br1;
            acc = __builtin_amdgcn_wmma_f32_16x16x4_f32(false, a, false, bm,
                                                        (short)0, acc, false, false);
        }
        // C/D layout: VGPR v -> M = v (lanes 0-15) or 8+v (lanes 16-31); N = lane&15
        const int nloc  = wc + (lane & 15);
        const int mbase = wr + ((lane >> 4) * 8);
        const float cn  = cs[nloc];
        #pragma unroll
        for (int v = 0; v < 8; ++v)
            sS[mbase + v][nloc] = acc[v] + cn;
    }
    __syncthreads();

    // ---------------- product phase: p = prod_d (1 - z^2), fused output ----------------
    {
        const int rb = (tid & 7) * 4;      // 4 consecutive rows
        const int cb = (tid >> 3) * 2;     // 2 consecutive cols
        float pr[4][2] = {{1.f,1.f},{1.f,1.f},{1.f,1.f},{1.f,1.f}};

        #pragma unroll 4
        for (int d = 0; d < Dsz; ++d) {
            const float4 xv = *(const float4*)&xsT[d][rb];   // ds_load_b128
            const float2 rv = *(const float2*)&rsT[d][cb];   // ds_load_b64
            const float2 bv = *(const float2*)&brT[d][cb];   // ds_load_b64
            const float xr[4] = {xv.x, xv.y, xv.z, xv.w};
            #pragma unroll
            for (int j = 0; j < 4; ++j) {
                const float z0 = fmaf(xr[j], rv.x, -bv.x);   // z = x*r - br
                const float t0 = z0 * z0;
                pr[j][0] = fmaf(-pr[j][0], t0, pr[j][0]);    // p *= (1 - z^2)
                const float z1 = fmaf(xr[j], rv.y, -bv.y);
                const float t1 = z1 * z1;
                pr[j][1] = fmaf(-pr[j][1], t1, pr[j][1]);
            }
        }
        #pragma unroll
        for (int j = 0; j < 4; ++j) {
            #pragma unroll
            for (int k = 0; k < 2; ++k) {
                const int m = rb + j, n = cb + k;
                const float s = sS[m][n];
                out[(size_t)(b0 + m) * Nsz + (n0 + n)] = pr[j][k] * __expf(-0.5f * s);
            }
        }
    }
}

extern "C" void kernel_launch(void* const* d_in, const int* in_sizes, int n_in,
                              void* d_out, int out_size, void* d_ws, size_t ws_size,
                              hipStream_t stream) {
    const float* x     = (const float*)d_in[0];   // (B, D)
    const float* bias  = (const float*)d_in[1];   // (N, D)
    const float* scale = (const float*)d_in[2];   // (N, D)
    float* out = (float*)d_out;                   // (B, N)
    (void)in_sizes; (void)n_in; (void)out_size; (void)d_ws; (void)ws_size;

    dim3 grid(Bsz / BT, Nsz / NT);   // 64 x 16 = 1024 workgroups
    dim3 block(128);                 // 4 wave32 waves
    hipLaunchKernelGGL(WaveletLayer02_68367289418120_kernel, grid, block, 0, stream,
                       x, bias, scale, out);
}